// CRF_87952340287730
// MI455X (gfx1250) — compile-verified
//
#include <hip/hip_runtime.h>

#define CRF_B 512
#define CRF_L 1024
#define CRF_T 128

// ---------------- CDNA5 async global->LDS path (guarded) ----------------
#if defined(__gfx1250__) && __has_builtin(__builtin_amdgcn_global_load_async_to_lds_b32)
#define CRF_HAVE_ASYNC 1
#else
#define CRF_HAVE_ASYNC 0
#endif

#if CRF_HAVE_ASYNC && __has_builtin(__builtin_amdgcn_global_load_async_to_lds_b128)
#define CRF_ASYNC_B128 1
#else
#define CRF_ASYNC_B128 0
#endif

#if CRF_HAVE_ASYNC
typedef __attribute__((address_space(1))) int crf_gas_int;  // global (AS1)
typedef __attribute__((address_space(3))) int crf_las_int;  // LDS (AS3)
#if CRF_ASYNC_B128
typedef int crf_v4i __attribute__((vector_size(4 * sizeof(int))));
typedef __attribute__((address_space(1))) crf_v4i crf_gas_v4i;  // global (AS1)
typedef __attribute__((address_space(3))) crf_v4i crf_las_v4i;  // LDS (AS3)
#endif
#endif

// Stage one 512B emit row (T floats) into LDS asynchronously (ASYNCcnt).
__device__ __forceinline__ void crf_stage_row(const float* grow, float* lrow, int j) {
#if CRF_HAVE_ASYNC
#if CRF_ASYNC_B128
  // 32 lanes x 16B async copies cover the row; 1/4 the issue slots of b32.
  if (j < 32) {
    __builtin_amdgcn_global_load_async_to_lds_b128(
        (crf_gas_v4i*)((unsigned long long)grow + 16ull * (unsigned)j),
        (crf_las_v4i*)(unsigned)((unsigned long long)lrow + 16u * (unsigned)j),
        0, 0);
  }
#else
  __builtin_amdgcn_global_load_async_to_lds_b32(
      (crf_gas_int*)(unsigned long long)(grow + j),
      (crf_las_int*)(unsigned)(unsigned long long)(lrow + j),
      0, 0);
#endif
#else
  lrow[j] = grow[j]; // synchronous fallback
#endif
}

#if CRF_HAVE_ASYNC
#if __has_builtin(__builtin_amdgcn_s_wait_asynccnt)
#define CRF_WAIT_ASYNC0() __builtin_amdgcn_s_wait_asynccnt(0)
#else
#define CRF_WAIT_ASYNC0() asm volatile("s_wait_asynccnt 0x0" ::: "memory")
#endif
#else
#define CRF_WAIT_ASYNC0() ((void)0)
#endif

// ---------------- Viterbi kernel: one workgroup per batch row ----------------
extern "C" __global__ void __launch_bounds__(CRF_T, 1)
crf_viterbi_kernel(const float* __restrict__ emit,   // [B, L, T]
                   const float* __restrict__ trans,  // [T, T]
                   const float* __restrict__ tagw,   // [T]
                   float* __restrict__ out)          // [B*L] path + [B] score
{
  // LDS: backpointers for the whole sequence stay on-chip (CDNA5: 320KB/WGP,
  // total here ~134KB -> 2 workgroups resident per WGP).
  __shared__ __align__(16) float prevbuf[2][CRF_T];          // 1 KB
  __shared__ __align__(16) float obsbuf[2][CRF_T];           // 1 KB
  __shared__ unsigned char bps[(CRF_L - 1) * CRF_T];         // 130,944 B
  __shared__ unsigned char pathsh[CRF_L];                    // 1 KB
  __shared__ float bestsh;

  const int j = threadIdx.x;      // tag owned by this lane
  const int b = blockIdx.x;       // batch row
  const float wj = tagw[j];
  const float* eb = emit + (size_t)b * CRF_L * CRF_T;

  // Transitions column j -> 128 VGPRs (one-time, coalesced across lanes).
  float tc[CRF_T];
#pragma unroll
  for (int i = 0; i < CRF_T; ++i) tc[i] = trans[i * CRF_T + j];

  // t = 0 init (emit scaled by tag_weight, matching reference).
  prevbuf[0][j] = eb[j] * wj;

  // Stage obs row 1, drain, publish to all waves.
  crf_stage_row(eb + 1 * CRF_T, &obsbuf[1][0], j);
  CRF_WAIT_ASYNC0();
  __syncthreads();

#pragma unroll 1
  for (int t = 1; t < CRF_L; ++t) {
    const int cur = (t - 1) & 1;
    const int nxt = t & 1;

    // Kick off row t+1 at the top; the ~512-instr compute body hides it.
    // Target buffer was last read at step t-1 (fenced by that step's barrier).
    if (t + 1 < CRF_L)
      crf_stage_row(eb + (size_t)(t + 1) * CRF_T, &obsbuf[(t + 1) & 1][0], j);

    // Row t landed before the previous barrier.
    const float obs = obsbuf[nxt][j] * wj;

    // max/argmax over i of prev[i] + trans[i][j]; prev read as LDS b128
    // broadcasts (uniform address, conflict-free), trans from registers.
    float best = -3.402823466e38f;
    int arg = 0;
    const float4* pv4 = reinterpret_cast<const float4*>(&prevbuf[cur][0]);
#pragma unroll
    for (int i4 = 0; i4 < CRF_T / 4; ++i4) {
      const float4 p = pv4[i4];
      float s;
      s = p.x + tc[4 * i4 + 0]; if (s > best) { best = s; arg = 4 * i4 + 0; }
      s = p.y + tc[4 * i4 + 1]; if (s > best) { best = s; arg = 4 * i4 + 1; }
      s = p.z + tc[4 * i4 + 2]; if (s > best) { best = s; arg = 4 * i4 + 2; }
      s = p.w + tc[4 * i4 + 3]; if (s > best) { best = s; arg = 4 * i4 + 3; }
    }

    bps[(t - 1) * CRF_T + j] = (unsigned char)arg;   // backpointer, LDS-resident
    prevbuf[nxt][j] = best + obs;

    // Drain row t+1, then publish prev + obs to all waves for step t+1.
    CRF_WAIT_ASYNC0();
    __syncthreads();
  }

  // Final argmax + backtrace, entirely from LDS (low-latency chase).
  if (j == 0) {
    const float* fin = &prevbuf[(CRF_L - 1) & 1][0];
    float bb = fin[0];
    int ba = 0;
    for (int i = 1; i < CRF_T; ++i) {
      const float v = fin[i];
      if (v > bb) { bb = v; ba = i; }
    }
    bestsh = bb;
    int c = ba;
    pathsh[CRF_L - 1] = (unsigned char)c;
    for (int t = CRF_L - 2; t >= 0; --t) {
      c = bps[t * CRF_T + c];
      pathsh[t] = (unsigned char)c;
    }
  }
  __syncthreads();

  // Cooperative, coalesced path write-out (path values as float).
#pragma unroll
  for (int k = 0; k < CRF_L / CRF_T; ++k) {
    const int idx = k * CRF_T + j;
    out[(size_t)b * CRF_L + idx] = (float)pathsh[idx];
  }
  if (j == 0) out[(size_t)CRF_B * CRF_L + b] = bestsh;
}

// ---------------- launch ----------------
extern "C" void kernel_launch(void* const* d_in, const int* in_sizes, int n_in,
                              void* d_out, int out_size, void* d_ws, size_t ws_size,
                              hipStream_t stream) {
  (void)in_sizes; (void)n_in; (void)out_size; (void)d_ws; (void)ws_size;
  const float* emit  = (const float*)d_in[0];   // [B, L, T] f32
  const float* trans = (const float*)d_in[1];   // [T, T]    f32
  const float* tagw  = (const float*)d_in[2];   // [T]       f32
  float* out = (float*)d_out;                   // [B*L] path + [B] score
  crf_viterbi_kernel<<<dim3(CRF_B), dim3(CRF_T), 0, stream>>>(emit, trans, tagw, out);
}